// MFA_13778255086332
// MI455X (gfx1250) — compile-verified
//
#include <hip/hip_runtime.h>
#include <math.h>

typedef __attribute__((ext_vector_type(2))) float v2f;
typedef __attribute__((ext_vector_type(8))) float v8f;
typedef unsigned int u32;
typedef __attribute__((ext_vector_type(4))) u32 v4u;
typedef __attribute__((ext_vector_type(8))) int v8i_;
typedef __attribute__((ext_vector_type(4))) int v4i_;

#define MM 32
#define NN 8192
#define DD 64
#define KK 16

// Per-component constant block in workspace (floats), contiguous so one TDM
// descriptor stages the whole thing into LDS.
// W and beta^T are stored K-PAIR INTERLEAVED: element (2p,c) and (2p+1,c)
// adjacent, so a WMMA B-fragment is one aligned 8-byte LDS read.
#define CB      5456            // block size (16-float aligned pad)
#define CB_W    0               // [32][64][2]  W = L^-T, pair-interleaved
#define CB_BT   4096            // [32][16][2]  beta^T,   pair-interleaved
#define CB_T1   5120            // [16][16]     term1
#define CB_MU   5376            // [64]         mu
#define CB_CO   5440            // [1]          log pi - 0.5(D log2pi + logdet)
#define CB_TL   5444            // elements moved by TDM (mult. of 4)

// ---------------- Kernel 1: per-component precompute (M=32 blocks, 64 thr) --
__global__ __launch_bounds__(64) void mfa_precompute(
    const float* __restrict__ pi, const float* __restrict__ muG,
    const float* __restrict__ LamG, const float* __restrict__ Psi,
    float* __restrict__ ws)
{
  __shared__ float S [64][64];   // Sigma, then Cholesky L in lower triangle
  __shared__ float Li[64][64];   // L^-1 (lower)
  __shared__ float Lam[64][16];
  __shared__ float T1 [64][16];  // Li @ Lam
  __shared__ float bT [64][16];  // beta^T
  __shared__ float s_logdet;

  const int m = blockIdx.x;
  const int t = threadIdx.x;     // 0..63
  float* blk = ws + (size_t)m * CB;

  for (int k = 0; k < 16; ++k) Lam[t][k] = LamG[(m*64 + t)*16 + k];
  __syncthreads();

  // Sigma row t = Lam[t,:] Lam^T + diag(Psi + 1e-6)
  for (int j = 0; j < 64; ++j) {
    float s = 0.f;
    for (int k = 0; k < 16; ++k) s += Lam[t][k] * Lam[j][k];
    if (j == t) s += Psi[t] + 1e-6f;
    S[t][j] = s;
  }
  __syncthreads();

  // Right-looking Cholesky, lower triangle in place
  for (int j = 0; j < 64; ++j) {
    if (t == 0) S[j][j] = sqrtf(S[j][j]);
    __syncthreads();
    if (t > j) S[t][j] /= S[j][j];
    __syncthreads();
    if (t > j) {
      float ltj = S[t][j];
      for (int c = j + 1; c <= t; ++c) S[t][c] -= ltj * S[c][j];
    }
    __syncthreads();
  }
  if (t == 0) {
    float ld = 0.f;
    for (int i = 0; i < 64; ++i) ld += logf(S[i][i]);
    s_logdet = 2.f * ld;
  }

  // Triangular inverse: thread t owns column t of Li
  Li[t][t] = 1.f / S[t][t];
  for (int i = t + 1; i < 64; ++i) {
    float s = 0.f;
    for (int k = t; k < i; ++k) s += S[i][k] * Li[k][t];
    Li[i][t] = -s / S[i][i];
  }
  for (int i = 0; i < t; ++i) Li[i][t] = 0.f;
  __syncthreads();

  // W row t = column t of Li, stored pair-interleaved:
  // blk[CB_W + ((t/2)*64 + j)*2 + (t&1)] = W[t][j]
  for (int j = 0; j < 64; ++j)
    blk[CB_W + (((t >> 1) * 64 + j) << 1) + (t & 1)] = Li[j][t];

  // T1 = Li @ Lam
  for (int k = 0; k < 16; ++k) {
    float s = 0.f;
    for (int j = 0; j <= t; ++j) s += Li[t][j] * Lam[j][k];
    T1[t][k] = s;
  }
  __syncthreads();
  // beta^T = Li^T @ T1, stored pair-interleaved like W
  for (int k = 0; k < 16; ++k) {
    float s = 0.f;
    for (int j = t; j < 64; ++j) s += Li[j][t] * T1[j][k];
    bT[t][k] = s;
    blk[CB_BT + (((t >> 1) * 16 + k) << 1) + (t & 1)] = s;
  }
  __syncthreads();
  // term1 = I - Lam^T @ beta^T  (4 entries per thread)
  for (int e = 0; e < 4; ++e) {
    int idx = t*4 + e, k1 = idx >> 4, k2 = idx & 15;
    float s = (k1 == k2) ? 1.f : 0.f;
    float acc = 0.f;
    for (int d = 0; d < 64; ++d) acc += bT[d][k1] * Lam[d][k2];
    blk[CB_T1 + idx] = s - acc;
  }
  blk[CB_MU + t] = muG[m*64 + t];
  if (t == 0)
    blk[CB_CO] = logf(pi[m] + 1e-10f)
               - 0.5f*(64.f*1.83787706640934548356f /*log(2pi)*/ + s_logdet);
}

// ---------------- Kernel 2: main per-(m, n-tile) WMMA kernel ----------------
// grid (N/64, M), 128 threads = 4 waves, each wave owns a 16-point row block.
// Per-m constants arrive via one Tensor Data Mover descriptor issued by wave 0
// while all waves prefetch their X fragments from global.
__global__ __launch_bounds__(128) void mfa_main(
    const float* __restrict__ X, const float* __restrict__ ws,
    float* __restrict__ out)
{
  __shared__ float sC[CB];          // W | beta^T | term1 | mu | coef
  __shared__ float sEz[4][16][16];

  const int m   = blockIdx.y;
  const int n0  = blockIdx.x * 64;
  const int tid = threadIdx.x;
  const int w   = tid >> 5;       // wave id 0..3
  const int l   = tid & 31;       // lane
  const int h   = l >> 4;         // half-wave
  const int r   = l & 15;         // A-row / B-col index
  const int n0w = n0 + w * 16;

  float (*sT1)[16] = (float(*)[16])(sC + CB_T1);
  float* sMu       = sC + CB_MU;

  // ---- TDM: stage the 5444-float constant block, wave 0 only -------------
  if (w == 0) {
    unsigned long long ga = (unsigned long long)(uintptr_t)(ws + (size_t)m * CB);
    u32 ldsa = (u32)(uintptr_t)&sC[0];   // low 32 bits of the generic address
    v4u g0;
    g0[0] = 1u;                                   // count=1, user mode
    g0[1] = ldsa;                                 // lds_addr
    g0[2] = (u32)ga;                              // global_addr[31:0]
    g0[3] = (u32)((ga >> 32) & 0x1FFFFFFull)      // global_addr[56:32]
          | (2u << 30);                           // type=2 ("image")
    v8i_ g1;
    g1[0] = (int)(2u << 16);                      // data_size=4B, no flags
    g1[1] = (int)((u32)CB_TL << 16);              // tensor_dim0[15:0]
    g1[2] = (int)(1u << 16);                      // tensor_dim1 = 1
    g1[3] = (int)((u32)CB_TL << 16);              // tile_dim0
    g1[4] = 0;                                    // tile_dim1/2 unused (1-D)
    g1[5] = CB_TL;                                // tensor_dim0_stride
    g1[6] = 0; g1[7] = 0;
    v4i_ gz4 = {0, 0, 0, 0};                      // groups 2/3: unused dims
    v8i_ gz8 = {0, 0, 0, 0, 0, 0, 0, 0};          // extra group (clang-23 form)
    __builtin_amdgcn_tensor_load_to_lds(g0, g1, gz4, gz4, gz8, 0);
    __builtin_amdgcn_s_wait_tensorcnt(0);
  }

  // ---- overlap: fetch X fragments (WMMA f32 A-layout) while TDM runs -----
  v2f xf[16];
  const float* Xrow = X + (size_t)(n0w + r) * 64;
#pragma unroll
  for (int k = 0; k < 16; ++k) {
    int c = 4*k + 2*h;
    xf[k].x = Xrow[c];
    xf[k].y = Xrow[c+1];
  }
  __syncthreads();

  // diff fragments: lane holds row r, cols 4k+2h+{0,1}
  v2f df[16];
#pragma unroll
  for (int k = 0; k < 16; ++k) {
    int c = 4*k + 2*h;
    df[k].x = xf[k].x - sMu[c];
    df[k].y = xf[k].y - sMu[c+1];
  }

  // r = diff @ W  (4 column tiles x 16 K-steps); maha = rowsum(r*r)
  // B-fragment (rows 4k+2h, 4k+2h+1 at col 16ct+r) = one aligned 8B LDS read
  float macc[8];
#pragma unroll
  for (int i = 0; i < 8; ++i) macc[i] = 0.f;
#pragma unroll
  for (int ct = 0; ct < 4; ++ct) {
    v8f acc = {};
#pragma unroll
    for (int k = 0; k < 16; ++k) {
      v2f bf = *(const v2f*)(sC + CB_W + ((((2*k + h) * 64) + 16*ct + r) << 1));
      acc = __builtin_amdgcn_wmma_f32_16x16x4_f32(false, df[k], false, bf,
                                                  (short)0, acc, false, false);
    }
#pragma unroll
    for (int i = 0; i < 8; ++i) macc[i] += acc[i]*acc[i];
  }

  // e_z = diff @ beta^T  (single 16x16 output tile, 16 K-steps)
  v8f ez = {};
#pragma unroll
  for (int k = 0; k < 16; ++k) {
    v2f bf = *(const v2f*)(sC + CB_BT + ((((2*k + h) * 16) + r) << 1));
    ez = __builtin_amdgcn_wmma_f32_16x16x4_f32(false, df[k], false, bf,
                                               (short)0, ez, false, false);
  }
#pragma unroll
  for (int i = 0; i < 8; ++i) sEz[w][i + 8*h][r] = ez[i];

  // reduce maha across the 16 lanes of each half (cols of the C tile)
#pragma unroll
  for (int i = 0; i < 8; ++i) {
    float v = macc[i];
    v += __shfl_xor(v, 1, 16);
    v += __shfl_xor(v, 2, 16);
    v += __shfl_xor(v, 4, 16);
    v += __shfl_xor(v, 8, 16);
    macc[i] = v;
  }
  if (r == 0) {
    float coef = sC[CB_CO];
#pragma unroll
    for (int i = 0; i < 8; ++i)   // C row = i + 8h  -> point n0w + i + 8h
      out[(size_t)(n0w + 8*h + i) * MM + m] = coef - 0.5f*macc[i];
  }
  __syncthreads();

  // write e_z [M,N,K], coalesced float4
  float* ezOut = out + (size_t)NN*MM;
#pragma unroll
  for (int rep = 0; rep < 2; ++rep) {
    int id4 = rep*32 + l;
    int p = id4 >> 2, c4 = (id4 & 3) << 2;
    float4 v = *(const float4*)&sEz[w][p][c4];
    *(float4*)(ezOut + ((size_t)m*NN + n0w + p)*16 + c4) = v;
  }

  // e_zz[n] = term1 + ez ez^T ; lane l writes float offsets l*4 and 128+l*4
  // of each point's 256-float block -> fully coalesced 512B stores per wave
  float* ezzOut = out + (size_t)NN*MM + (size_t)MM*NN*KK;
  const int i0 = l >> 2;
  const int j0 = (l & 3) << 2;
  float4 t1a = *(const float4*)&sT1[i0][j0];
  float4 t1b = *(const float4*)&sT1[i0+8][j0];
#pragma unroll
  for (int p = 0; p < 16; ++p) {
    float4 ej = *(const float4*)&sEz[w][p][j0];
    float ei  = sEz[w][p][i0];
    float ei2 = sEz[w][p][i0+8];
    float4 o1 = make_float4(t1a.x + ei*ej.x,  t1a.y + ei*ej.y,
                            t1a.z + ei*ej.z,  t1a.w + ei*ej.w);
    float4 o2 = make_float4(t1b.x + ei2*ej.x, t1b.y + ei2*ej.y,
                            t1b.z + ei2*ej.z, t1b.w + ei2*ej.w);
    float* base = ezzOut + ((size_t)m*NN + n0w + p) * 256;
    *(float4*)(base + l*4)       = o1;   // rows 0..7  (i0*16 + j0 == l*4)
    *(float4*)(base + 128 + l*4) = o2;   // rows 8..15
  }
}

// ---------------- Kernel 3: in-place softmax over M per point ---------------
__global__ __launch_bounds__(256) void mfa_softmax(float* __restrict__ out)
{
  int n = blockIdx.x * 256 + threadIdx.x;
  float* row = out + (size_t)n * MM;
  float v[MM];
  float mx = -3.4e38f;
#pragma unroll
  for (int i = 0; i < MM; ++i) { v[i] = row[i]; mx = fmaxf(mx, v[i]); }
  float s = 0.f;
#pragma unroll
  for (int i = 0; i < MM; ++i) { v[i] = __expf(v[i] - mx); s += v[i]; }
  float inv = 1.f / s;
#pragma unroll
  for (int i = 0; i < MM; ++i) row[i] = v[i] * inv;
}

extern "C" void kernel_launch(void* const* d_in, const int* in_sizes, int n_in,
                              void* d_out, int out_size, void* d_ws, size_t ws_size,
                              hipStream_t stream) {
  const float* X   = (const float*)d_in[0];
  const float* pi  = (const float*)d_in[1];
  const float* mu  = (const float*)d_in[2];
  const float* Lam = (const float*)d_in[3];
  const float* Psi = (const float*)d_in[4];
  float* out = (float*)d_out;
  float* ws  = (float*)d_ws;   // needs 32*5456*4 = ~698 KB

  mfa_precompute<<<MM, 64, 0, stream>>>(pi, mu, Lam, Psi, ws);
  dim3 g2(NN/64, MM);
  mfa_main<<<g2, 128, 0, stream>>>(X, ws, out);
  mfa_softmax<<<NN/256, 256, 0, stream>>>(out);
}